// GCNStudent_42494406427536
// MI455X (gfx1250) — compile-verified
//
#include <hip/hip_runtime.h>
#include <hip/hip_bf16.h>

typedef __attribute__((ext_vector_type(2))) float v2f;
typedef __attribute__((ext_vector_type(8))) float v8f;

#define IN_FEATS 128
#define NUM_CLASSES 16

// ---------------------------------------------------------------------------
// 1) Zero accumulators: d_out (N*16 floats) and both degree arrays (N u32).
// ---------------------------------------------------------------------------
__global__ void gcn_zero_kernel(float* __restrict__ out,
                                unsigned* __restrict__ deg_out,
                                unsigned* __restrict__ deg_in,
                                int n_nodes) {
    int idx = blockIdx.x * blockDim.x + threadIdx.x;
    int total = n_nodes * NUM_CLASSES;
    if (idx < total) out[idx] = 0.0f;
    if (idx < n_nodes) { deg_out[idx] = 0u; deg_in[idx] = 0u; }
}

// ---------------------------------------------------------------------------
// 2) Degree counting: one thread per edge, u32 atomics.
// ---------------------------------------------------------------------------
__global__ void gcn_degree_kernel(const int* __restrict__ src,
                                  const int* __restrict__ dst,
                                  unsigned* __restrict__ deg_out,
                                  unsigned* __restrict__ deg_in,
                                  int n_edges) {
    int e = blockIdx.x * blockDim.x + threadIdx.x;
    if (e < n_edges) {
        atomicAdd(&deg_out[src[e]], 1u);
        atomicAdd(&deg_in[dst[e]], 1u);
    }
}

// ---------------------------------------------------------------------------
// 3) XW GEMM via V_WMMA_F32_16X16X4_F32, fused with norm_src scaling.
//    One wave32 computes one 16x16 tile: 16 nodes x 16 classes, K = 128.
//    A (16x4 f32, 2 VGPR):  lane L -> M = L%16, Kpair = 2*(L/16)
//    B (4x16 f32, 2 VGPR):  lane L -> N = L%16, Kpair = 2*(L/16)
//    C/D (16x16 f32, 8 VGPR): VGPR r, lanes0-15 -> M=r, lanes16-31 -> M=r+8,
//                             N = L%16.
// ---------------------------------------------------------------------------
__global__ void gcn_xw_wmma_kernel(const float* __restrict__ X,
                                   const float* __restrict__ W,
                                   const unsigned* __restrict__ deg_out,
                                   float* __restrict__ xw_scaled,
                                   int n_nodes) {
    int wave = (blockIdx.x * blockDim.x + threadIdx.x) >> 5;
    int lane = threadIdx.x & 31;
    int tile = wave;
    if (tile * 16 >= n_nodes) return;   // whole wave exits together

    int mn   = lane & 15;   // row of A / col of B / col of C
    int half = lane >> 4;   // selects K pair within 4-wide K step

    const float* __restrict__ xrow = X + (size_t)(tile * 16 + mn) * IN_FEATS;

    v8f c = {};
    #pragma unroll
    for (int k0 = 0; k0 < IN_FEATS; k0 += 4) {
        int k = k0 + 2 * half;
        v2f a, b;
        a.x = xrow[k + 0];
        a.y = xrow[k + 1];
        b.x = W[(k + 0) * NUM_CLASSES + mn];
        b.y = W[(k + 1) * NUM_CLASSES + mn];
        // 8 args: (neg_a, A, neg_b, B, c_mod, C, reuse_a, reuse_b)
        c = __builtin_amdgcn_wmma_f32_16x16x4_f32(false, a, false, b,
                                                  (short)0, c, false, false);
    }

    // Scale each output row by norm_src = rsqrt(deg_out) and store.
    #pragma unroll
    for (int r = 0; r < 8; ++r) {
        int m = tile * 16 + r + 8 * half;
        unsigned d = deg_out[m];
        float ns = (d > 0u) ? rsqrtf((float)d) : 0.0f;
        xw_scaled[(size_t)m * NUM_CLASSES + mn] = c[r] * ns;
    }
}

// ---------------------------------------------------------------------------
// 4) Edge scatter: 4 threads per edge, each moves 4 consecutive classes.
//    Gather is a coalesced float4 (4 threads -> 64B contiguous per edge),
//    accumulate with global_atomic_add_f32 into d_out.
// ---------------------------------------------------------------------------
__global__ void gcn_scatter_kernel(const int* __restrict__ src,
                                   const int* __restrict__ dst,
                                   const float* __restrict__ xw_scaled,
                                   float* __restrict__ out,
                                   int n_edges) {
    int tid  = blockIdx.x * blockDim.x + threadIdx.x;
    int e    = tid >> 2;
    int part = tid & 3;
    if (e >= n_edges) return;

    int s = src[e];
    int d = dst[e];

    const float4* g = (const float4*)(xw_scaled + (size_t)s * NUM_CLASSES);
    float4 v = g[part];

    float* o = out + (size_t)d * NUM_CLASSES + part * 4;
    atomicAdd(&o[0], v.x);
    atomicAdd(&o[1], v.y);
    atomicAdd(&o[2], v.z);
    atomicAdd(&o[3], v.w);
}

// ---------------------------------------------------------------------------
// 5) Epilogue: h = relu(agg * rsqrt(deg_in) + b), in place on d_out.
// ---------------------------------------------------------------------------
__global__ void gcn_epilogue_kernel(float* __restrict__ out,
                                    const unsigned* __restrict__ deg_in,
                                    const float* __restrict__ bias,
                                    int n_nodes) {
    int idx = blockIdx.x * blockDim.x + threadIdx.x;
    int total = n_nodes * NUM_CLASSES;
    if (idx >= total) return;
    int n = idx >> 4;          // /16
    int cidx = idx & 15;
    unsigned d = deg_in[n];
    float nd = (d > 0u) ? rsqrtf((float)d) : 0.0f;
    float v = out[idx] * nd + bias[cidx];
    out[idx] = v > 0.0f ? v : 0.0f;
}

extern "C" void kernel_launch(void* const* d_in, const int* in_sizes, int n_in,
                              void* d_out, int out_size, void* d_ws, size_t ws_size,
                              hipStream_t stream) {
    const float* in_feat = (const float*)d_in[0];
    const int*   src     = (const int*)d_in[1];
    const int*   dst     = (const int*)d_in[2];
    const float* W       = (const float*)d_in[3];
    const float* b       = (const float*)d_in[4];
    float* out = (float*)d_out;

    int n_nodes = in_sizes[0] / IN_FEATS;
    int n_edges = in_sizes[1];

    // Workspace layout: [deg_out u32 N][deg_in u32 N][xw_scaled f32 N*16]
    unsigned* deg_out   = (unsigned*)d_ws;
    unsigned* deg_in    = deg_out + n_nodes;
    float*    xw_scaled = (float*)(deg_in + n_nodes);

    const int BLK = 256;

    // 1) zero accumulators
    {
        int total = n_nodes * NUM_CLASSES;
        gcn_zero_kernel<<<(total + BLK - 1) / BLK, BLK, 0, stream>>>(
            out, deg_out, deg_in, n_nodes);
    }
    // 2) degrees
    gcn_degree_kernel<<<(n_edges + BLK - 1) / BLK, BLK, 0, stream>>>(
        src, dst, deg_out, deg_in, n_edges);
    // 3) XW via WMMA (one wave per 16-row tile)
    {
        int n_tiles  = (n_nodes + 15) / 16;
        int n_threads = n_tiles * 32;
        gcn_xw_wmma_kernel<<<(n_threads + BLK - 1) / BLK, BLK, 0, stream>>>(
            in_feat, W, deg_out, xw_scaled, n_nodes);
    }
    // 4) edge gather + scatter-add
    {
        long long n_threads = (long long)n_edges * 4;
        gcn_scatter_kernel<<<(int)((n_threads + BLK - 1) / BLK), BLK, 0, stream>>>(
            src, dst, xw_scaled, out, n_edges);
    }
    // 5) epilogue
    {
        int total = n_nodes * NUM_CLASSES;
        gcn_epilogue_kernel<<<(total + BLK - 1) / BLK, BLK, 0, stream>>>(
            out, deg_in, b, n_nodes);
    }
}